// LongContextAttention_26182120636575
// MI455X (gfx1250) — compile-verified
//
#include <hip/hip_runtime.h>
#include <hip/hip_bf16.h>
#include <stdint.h>

typedef __attribute__((ext_vector_type(2))) float v2f;
typedef __attribute__((ext_vector_type(8))) float v8f;
typedef __attribute__((ext_vector_type(4))) unsigned int v4u;
typedef __attribute__((ext_vector_type(8))) unsigned int v8u;

#define B_SZ   2
#define L_SEQ  2048
#define D_MOD  1024
#define N_HEAD 16
#define HDIM   64
#define M_ROWS (B_SZ * L_SEQ)   // 4096
#define NLM    32               // landmarks

// LDS byte offset of a __shared__ object: flat shared pointers carry the LDS
// address in their low 32 bits (ISA 10.2 aperture mapping).
__device__ __forceinline__ uint32_t lds_off(const void* p) {
    return (uint32_t)(uintptr_t)p;
}

// ---------------------------------------------------------------------------
// Tensor Data Mover: DMA a 2-D tile (tile_w x tile_h elements of f32, row
// stride stride_elts) from global memory into LDS at lds_addr (packed rows).
// Issue from ONE wave only; wait with s_wait_tensorcnt before the barrier.
// D# layout per CDNA5 ISA.
// ---------------------------------------------------------------------------
__device__ __forceinline__ void tdm_load_2d(uint32_t lds_addr, const void* gptr,
                                            uint32_t tile_w, uint32_t tile_h,
                                            uint32_t stride_elts) {
    uint64_t ga = (uint64_t)(uintptr_t)gptr;
    v4u g0;
    g0[0] = 1u;                                    // count=1 (valid user D#)
    g0[1] = lds_addr;                              // lds_addr
    g0[2] = (uint32_t)ga;                          // global_addr[31:0]
    g0[3] = ((uint32_t)(ga >> 32) & 0x01FFFFFFu)   // global_addr[56:32]
            | 0x80000000u;                         // type=2 ("image")
    v8u g1;
    g1[0] = 0x00020000u;                           // data_size=2 (4 bytes)
    g1[1] = (tile_w & 0xFFFFu) << 16;              // tensor_dim0[15:0]
    g1[2] = (tile_h & 0xFFFFu) << 16;              // tensor_dim0 hi | tensor_dim1 lo
    g1[3] = (tile_w & 0xFFFFu) << 16;              // tensor_dim1 hi | tile_dim0
    g1[4] = (tile_h & 0xFFFFu);                    // tile_dim1 | tile_dim2=0
    g1[5] = stride_elts;                           // tensor_dim0_stride[31:0]
    g1[6] = 0u;                                    // stride hi | dim1_stride lo
    g1[7] = 0u;
    asm volatile("tensor_load_to_lds %0, %1" :: "s"(g0), "s"(g1) : "memory");
}

// Per-lane async copy of 16B global -> LDS (ASYNCcnt).
__device__ __forceinline__ void async_b128(uint32_t lds_addr, const void* gptr) {
    asm volatile("global_load_async_to_lds_b128 %0, %1, off"
                 :: "v"(lds_addr), "v"(gptr) : "memory");
}
__device__ __forceinline__ void wait_async0() {
    asm volatile("s_wait_asynccnt 0x0" ::: "memory");
}

// ---------------------------------------------------------------------------
// Kernel 1: fused Q/K/V projection.  Y = X @ W^T, scattered to (b,h,l,d).
// Block = 128 threads (4 waves), tile 64(M) x 64(N), K chunked by 32 via TDM.
// ---------------------------------------------------------------------------
__global__ __launch_bounds__(128)
void lca_qkv_proj(const float* __restrict__ x,
                  const float* __restrict__ wq,
                  const float* __restrict__ wk,
                  const float* __restrict__ wv,
                  float* __restrict__ qb,
                  float* __restrict__ kb,
                  float* __restrict__ vb) {
    __shared__ float As[64][32];
    __shared__ float Bs[64][32];

    const int t    = threadIdx.x;
    const int wave = t >> 5;
    const int lane = t & 31;
    const int ln   = lane & 15;
    const int hi   = lane >> 4;
    const int m0   = blockIdx.x * 64;
    const int n0   = blockIdx.y * 64;

    const float* W   = (blockIdx.z == 0) ? wq : (blockIdx.z == 1) ? wk : wv;
    float*       dst = (blockIdx.z == 0) ? qb : (blockIdx.z == 1) ? kb : vb;

    const uint32_t as_off = lds_off(&As[0][0]);
    const uint32_t bs_off = lds_off(&Bs[0][0]);

    v8f acc[4];
#pragma unroll
    for (int i = 0; i < 4; ++i)
#pragma unroll
        for (int j = 0; j < 8; ++j) acc[i][j] = 0.0f;

    for (int kc = 0; kc < D_MOD; kc += 32) {
        __syncthreads();
        if (wave == 0) {
            tdm_load_2d(as_off, &x[(size_t)m0 * D_MOD + kc], 32, 64, D_MOD);
            tdm_load_2d(bs_off, &W[(size_t)n0 * D_MOD + kc], 32, 64, D_MOD);
            __builtin_amdgcn_s_wait_tensorcnt(0);
        }
        __syncthreads();
#pragma unroll
        for (int kk = 0; kk < 8; ++kk) {
            const int kb4 = kk * 4 + hi * 2;
            v2f a = *(const v2f*)&As[wave * 16 + ln][kb4];
#pragma unroll
            for (int tt = 0; tt < 4; ++tt) {
                v2f b = *(const v2f*)&Bs[tt * 16 + ln][kb4];
                acc[tt] = __builtin_amdgcn_wmma_f32_16x16x4_f32(
                    false, a, false, b, (short)0, acc[tt], false, false);
            }
        }
    }

#pragma unroll
    for (int tt = 0; tt < 4; ++tt)
#pragma unroll
        for (int r = 0; r < 8; ++r) {
            int m  = m0 + wave * 16 + r + hi * 8;
            int n  = n0 + tt * 16 + ln;
            int bi = m >> 11, l = m & 2047;
            int h  = n >> 6,  dd = n & 63;
            dst[((size_t)(bi * N_HEAD + h) * L_SEQ + l) * HDIM + dd] = acc[tt][r];
        }
}

// ---------------------------------------------------------------------------
// Kernel 2: flash attention + linear-memory read, per (b*h, q-tile of 64).
// ---------------------------------------------------------------------------
__global__ __launch_bounds__(128)
void lca_attention(const float* __restrict__ qbase,
                   const float* __restrict__ kbase,
                   const float* __restrict__ vbase,
                   const float* __restrict__ fcos,
                   const float* __restrict__ fsin,
                   const float* __restrict__ Mold,
                   const float* __restrict__ Zold,
                   const float* __restrict__ mgate,
                   float* __restrict__ attn_out) {
    __shared__ float qs[64][64];      // roped Q
    __shared__ float ks[64][64];      // roped K   (phase B: phi(Q))
    __shared__ float vs[64][64];      // V         (phase B: M_old)
    __shared__ float ps[4][16][64];   // per-wave P staging (phase B: Z cache)

    const int t    = threadIdx.x;
    const int wave = t >> 5;
    const int lane = t & 31;
    const int ln   = lane & 15;
    const int hi   = lane >> 4;
    const int bh   = blockIdx.x;      // 0..31
    const int qt   = blockIdx.y;      // 0..31

    const uint32_t vs_off = lds_off(&vs[0][0]);

    // ---- load Q tile with RoPE ----
#pragma unroll
    for (int i = 0; i < 16; ++i) {
        int p   = t + 128 * i;        // 2048 pairs
        int row = p >> 5;
        int jd  = p & 31;
        int gr  = qt * 64 + row;
        float2 qv = *(const float2*)&qbase[((size_t)bh * L_SEQ + gr) * HDIM + 2 * jd];
        float c = fcos[gr * 32 + jd];
        float s = fsin[gr * 32 + jd];
        qs[row][2 * jd]     = qv.x * c - qv.y * s;
        qs[row][2 * jd + 1] = qv.x * s + qv.y * c;
    }

    v8f Oacc[4];
    float mrow[8], lrow[8];
#pragma unroll
    for (int i = 0; i < 4; ++i)
#pragma unroll
        for (int j = 0; j < 8; ++j) Oacc[i][j] = 0.0f;
#pragma unroll
    for (int r = 0; r < 8; ++r) { mrow[r] = -3.0e38f; lrow[r] = 0.0f; }

    for (int kt = 0; kt <= qt; ++kt) {
        __syncthreads();
        // V tile: per-lane async copy to LDS (contiguous 4KB block, rows 0..15)
#pragma unroll
        for (int i = 0; i < 2; ++i) {
            int f   = t + 128 * i;    // 256 x float4 slots
            int row = f >> 4;
            int c4  = (f & 15) << 2;
            async_b128(lds_off(&vs[row][c4]),
                       &vbase[((size_t)bh * L_SEQ + kt * 64 + row) * HDIM + c4]);
        }
        // K tile (roped), regular path (needs elementwise math)
#pragma unroll
        for (int i = 0; i < 16; ++i) {
            int p   = t + 128 * i;
            int row = p >> 5;
            int jd  = p & 31;
            int gr  = kt * 64 + row;
            float2 kv = *(const float2*)&kbase[((size_t)bh * L_SEQ + gr) * HDIM + 2 * jd];
            float c = fcos[gr * 32 + jd];
            float s = fsin[gr * 32 + jd];
            ks[row][2 * jd]     = kv.x * c - kv.y * s;
            ks[row][2 * jd + 1] = kv.x * s + kv.y * c;
        }
        wait_async0();
        __syncthreads();

        // S = Q . K^T
        v8f Sacc[4];
#pragma unroll
        for (int i = 0; i < 4; ++i)
#pragma unroll
            for (int j = 0; j < 8; ++j) Sacc[i][j] = 0.0f;
#pragma unroll
        for (int kk = 0; kk < 16; ++kk) {
            const int kb4 = kk * 4 + hi * 2;
            v2f a = *(const v2f*)&qs[wave * 16 + ln][kb4];
#pragma unroll
            for (int tt = 0; tt < 4; ++tt) {
                v2f b = *(const v2f*)&ks[tt * 16 + ln][kb4];
                Sacc[tt] = __builtin_amdgcn_wmma_f32_16x16x4_f32(
                    false, a, false, b, (short)0, Sacc[tt], false, false);
            }
        }

        // scale + causal mask
#pragma unroll
        for (int tt = 0; tt < 4; ++tt)
#pragma unroll
            for (int r = 0; r < 8; ++r) {
                int qrow = qt * 64 + wave * 16 + r + hi * 8;
                int kcol = kt * 64 + tt * 16 + ln;
                float sv = Sacc[tt][r] * 0.125f;
                Sacc[tt][r] = (kcol > qrow) ? -3.0e38f : sv;
            }

        // online softmax (row lives in a 16-lane half of the wave)
#pragma unroll
        for (int r = 0; r < 8; ++r) {
            float mx = Sacc[0][r];
#pragma unroll
            for (int tt = 1; tt < 4; ++tt) mx = fmaxf(mx, Sacc[tt][r]);
#pragma unroll
            for (int off = 1; off < 16; off <<= 1)
                mx = fmaxf(mx, __shfl_xor(mx, off, 32));
            float mnew  = fmaxf(mrow[r], mx);
            float alpha = __expf(mrow[r] - mnew);
            float psum  = 0.0f;
#pragma unroll
            for (int tt = 0; tt < 4; ++tt) {
                float pv = __expf(Sacc[tt][r] - mnew);
                Sacc[tt][r] = pv;
                psum += pv;
            }
#pragma unroll
            for (int off = 1; off < 16; off <<= 1)
                psum += __shfl_xor(psum, off, 32);
            lrow[r] = lrow[r] * alpha + psum;
            mrow[r] = mnew;
#pragma unroll
            for (int tt = 0; tt < 4; ++tt) Oacc[tt][r] *= alpha;
        }

        // stage P (C-layout -> LDS row-major), per-wave region
#pragma unroll
        for (int tt = 0; tt < 4; ++tt)
#pragma unroll
            for (int r = 0; r < 8; ++r)
                ps[wave][r + 8 * hi][tt * 16 + ln] = Sacc[tt][r];

        // O += P . V
#pragma unroll
        for (int kk = 0; kk < 16; ++kk) {
            const int kb4 = kk * 4 + hi * 2;
            v2f a = *(const v2f*)&ps[wave][ln][kb4];
#pragma unroll
            for (int tt = 0; tt < 4; ++tt) {
                v2f b;
                b[0] = vs[kb4][tt * 16 + ln];
                b[1] = vs[kb4 + 1][tt * 16 + ln];
                Oacc[tt] = __builtin_amdgcn_wmma_f32_16x16x4_f32(
                    false, a, false, b, (short)0, Oacc[tt], false, false);
            }
        }
    }

    // normalize
#pragma unroll
    for (int r = 0; r < 8; ++r) {
        float inv = 1.0f / lrow[r];
#pragma unroll
        for (int tt = 0; tt < 4; ++tt) Oacc[tt][r] *= inv;
    }

    // ---- phase B: linear-memory read: gate * (phi(Q) M_old) / (phi(Q).Z) ----
    __syncthreads();
    float* zcache = &ps[0][0][0];
#pragma unroll
    for (int i = 0; i < 32; ++i) {
        int f   = t + 128 * i;       // 4096 elements
        int row = f >> 6;
        int dd  = f & 63;
        float qv = qbase[((size_t)bh * L_SEQ + qt * 64 + row) * HDIM + dd];
        ks[row][dd] = (qv > 0.0f) ? qv + 1.0f : __expf(qv);   // elu + 1
    }
    if (wave == 0) {
        tdm_load_2d(vs_off, &Mold[(size_t)bh * 4096], 64, 64, 64);  // M_old tile
        __builtin_amdgcn_s_wait_tensorcnt(0);
    }
    if (t < 64) zcache[t] = Zold[bh * 64 + t];
    __syncthreads();

    v8f Nacc[4];
#pragma unroll
    for (int i = 0; i < 4; ++i)
#pragma unroll
        for (int j = 0; j < 8; ++j) Nacc[i][j] = 0.0f;
#pragma unroll
    for (int kk = 0; kk < 16; ++kk) {
        const int kb4 = kk * 4 + hi * 2;
        v2f a = *(const v2f*)&ks[wave * 16 + ln][kb4];
#pragma unroll
        for (int tt = 0; tt < 4; ++tt) {
            v2f b;
            b[0] = vs[kb4][tt * 16 + ln];
            b[1] = vs[kb4 + 1][tt * 16 + ln];
            Nacc[tt] = __builtin_amdgcn_wmma_f32_16x16x4_f32(
                false, a, false, b, (short)0, Nacc[tt], false, false);
        }
    }

    float den[8];
#pragma unroll
    for (int r = 0; r < 8; ++r) {
        int row = wave * 16 + r + 8 * hi;
        float d = 1e-6f;
        for (int dd = 0; dd < 64; ++dd) d += ks[row][dd] * zcache[dd];
        den[r] = d;
    }
    float gate = 1.0f / (1.0f + __expf(-mgate[0]));

#pragma unroll
    for (int tt = 0; tt < 4; ++tt)
#pragma unroll
        for (int r = 0; r < 8; ++r) {
            int row = qt * 64 + wave * 16 + r + 8 * hi;
            int col = tt * 16 + ln;
            attn_out[((size_t)bh * L_SEQ + row) * HDIM + col] =
                Oacc[tt][r] + gate * Nacc[tt][r] / den[r];
        }
}

// ---------------------------------------------------------------------------
// Kernel 3: landmark compression + memory update (WMMA for delta_M).
// ---------------------------------------------------------------------------
__global__ __launch_bounds__(128)
void lca_landmarks(const float* __restrict__ kbase,
                   const float* __restrict__ vbase,
                   const float* __restrict__ Mold,
                   const float* __restrict__ Zold,
                   const float* __restrict__ mdecay,
                   float* __restrict__ Mout,
                   float* __restrict__ Zout) {
    __shared__ float pk[NLM][64];   // phi(landmark K)
    __shared__ float lv[NLM][64];   // landmark V
    const int bh   = blockIdx.x;
    const int t    = threadIdx.x;
    const int wave = t >> 5;
    const int lane = t & 31;
    const int ln   = lane & 15;
    const int hi   = lane >> 4;

#pragma unroll
    for (int i = 0; i < 16; ++i) {
        int f = t + 128 * i;          // 0..2047
        int m = f >> 6, dd = f & 63;
        const float* kr = &kbase[((size_t)bh * L_SEQ + m * 64) * HDIM + dd];
        const float* vr = &vbase[((size_t)bh * L_SEQ + m * 64) * HDIM + dd];
        float sk = 0.0f, sv = 0.0f;
        for (int rr = 0; rr < 64; ++rr) { sk += kr[rr * 64]; sv += vr[rr * 64]; }
        sk *= (1.0f / 64.0f);
        sv *= (1.0f / 64.0f);
        pk[m][dd] = (sk > 0.0f) ? sk + 1.0f : __expf(sk);   // elu + 1
        lv[m][dd] = sv;
    }
    __syncthreads();

    // delta_M[d][e] = sum_m pk[m][d] * lv[m][e]   (A = pk^T, K=32)
    v8f acc[4];
#pragma unroll
    for (int i = 0; i < 4; ++i)
#pragma unroll
        for (int j = 0; j < 8; ++j) acc[i][j] = 0.0f;
#pragma unroll
    for (int kk = 0; kk < 8; ++kk) {
        const int kb4 = kk * 4 + hi * 2;
        v2f a;
        a[0] = pk[kb4][wave * 16 + ln];
        a[1] = pk[kb4 + 1][wave * 16 + ln];
#pragma unroll
        for (int tt = 0; tt < 4; ++tt) {
            v2f b;
            b[0] = lv[kb4][tt * 16 + ln];
            b[1] = lv[kb4 + 1][tt * 16 + ln];
            acc[tt] = __builtin_amdgcn_wmma_f32_16x16x4_f32(
                false, a, false, b, (short)0, acc[tt], false, false);
        }
    }

    float decay = 1.0f / (1.0f + __expf(-mdecay[0]));
#pragma unroll
    for (int tt = 0; tt < 4; ++tt)
#pragma unroll
        for (int r = 0; r < 8; ++r) {
            int d = wave * 16 + r + 8 * hi;
            int e = tt * 16 + ln;
            int idx = d * 64 + e;
            Mout[(size_t)bh * 4096 + idx] =
                Mold[(size_t)bh * 4096 + idx] * decay + acc[tt][r];
        }
    if (t < 64) {
        float s = 0.0f;
        for (int m = 0; m < NLM; ++m) s += pk[m][t];
        Zout[bh * 64 + t] = Zold[bh * 64 + t] * decay + s;
    }
}

// ---------------------------------------------------------------------------
// Kernel 4: output projection  y = attn_out(head-merged) @ wo^T
// ---------------------------------------------------------------------------
__global__ __launch_bounds__(128)
void lca_out_proj(const float* __restrict__ ao,
                  const float* __restrict__ wo,
                  float* __restrict__ y) {
    __shared__ float As[64][32];
    __shared__ float Bs[64][32];

    const int t    = threadIdx.x;
    const int wave = t >> 5;
    const int lane = t & 31;
    const int ln   = lane & 15;
    const int hi   = lane >> 4;
    const int m0   = blockIdx.x * 64;
    const int n0   = blockIdx.y * 64;

    const uint32_t as_off = lds_off(&As[0][0]);
    const uint32_t bs_off = lds_off(&Bs[0][0]);

    v8f acc[4];
#pragma unroll
    for (int i = 0; i < 4; ++i)
#pragma unroll
        for (int j = 0; j < 8; ++j) acc[i][j] = 0.0f;

    const int bi = m0 >> 11;          // batch of this M tile
    const int l0 = m0 & 2047;         // sequence base of this M tile

    for (int kc = 0; kc < D_MOD; kc += 32) {
        __syncthreads();
        if (wave == 0) {
            int h   = kc >> 6;        // head of this K chunk
            int dd0 = kc & 63;        // offset within head
            tdm_load_2d(as_off,
                        &ao[((size_t)(bi * N_HEAD + h) * L_SEQ + l0) * HDIM + dd0],
                        32, 64, HDIM);
            tdm_load_2d(bs_off, &wo[(size_t)n0 * D_MOD + kc], 32, 64, D_MOD);
            __builtin_amdgcn_s_wait_tensorcnt(0);
        }
        __syncthreads();
#pragma unroll
        for (int kk = 0; kk < 8; ++kk) {
            const int kb4 = kk * 4 + hi * 2;
            v2f a = *(const v2f*)&As[wave * 16 + ln][kb4];
#pragma unroll
            for (int tt = 0; tt < 4; ++tt) {
                v2f b = *(const v2f*)&Bs[tt * 16 + ln][kb4];
                acc[tt] = __builtin_amdgcn_wmma_f32_16x16x4_f32(
                    false, a, false, b, (short)0, acc[tt], false, false);
            }
        }
    }

#pragma unroll
    for (int tt = 0; tt < 4; ++tt)
#pragma unroll
        for (int r = 0; r < 8; ++r) {
            int m = m0 + wave * 16 + r + hi * 8;
            int n = n0 + tt * 16 + ln;
            y[(size_t)m * D_MOD + n] = acc[tt][r];
        }
}

// ---------------------------------------------------------------------------
extern "C" void kernel_launch(void* const* d_in, const int* in_sizes, int n_in,
                              void* d_out, int out_size, void* d_ws, size_t ws_size,
                              hipStream_t stream) {
    const float* x      = (const float*)d_in[0];
    const float* wq     = (const float*)d_in[1];
    const float* wk     = (const float*)d_in[2];
    const float* wv     = (const float*)d_in[3];
    const float* wo     = (const float*)d_in[4];
    const float* mgate  = (const float*)d_in[5];
    const float* mdecay = (const float*)d_in[6];
    const float* fcos   = (const float*)d_in[7];
    const float* fsin   = (const float*)d_in[8];
    const float* Mold   = (const float*)d_in[9];
    const float* Zold   = (const float*)d_in[10];

    float* out  = (float*)d_out;
    float* y    = out;                                  // (B, L, D)
    float* Mnew = out + (size_t)B_SZ * L_SEQ * D_MOD;   // (B, H, hd, hd)
    float* Znew = Mnew + (size_t)B_SZ * N_HEAD * HDIM * HDIM;

    float* ws = (float*)d_ws;
    const size_t tsz = (size_t)B_SZ * N_HEAD * L_SEQ * HDIM;  // 4M floats
    float* qb = ws;
    float* kb = ws + tsz;
    float* vb = ws + 2 * tsz;
    float* ao = ws + 3 * tsz;

    dim3 gProj(M_ROWS / 64, D_MOD / 64, 3);
    lca_qkv_proj<<<gProj, dim3(128), 0, stream>>>(x, wq, wk, wv, qb, kb, vb);

    dim3 gAttn(B_SZ * N_HEAD, L_SEQ / 64);
    lca_attention<<<gAttn, dim3(128), 0, stream>>>(qb, kb, vb, fcos, fsin,
                                                   Mold, Zold, mgate, ao);

    lca_landmarks<<<dim3(B_SZ * N_HEAD), dim3(128), 0, stream>>>(
        kb, vb, Mold, Zold, mdecay, Mnew, Znew);

    dim3 gOut(M_ROWS / 64, D_MOD / 64);
    lca_out_proj<<<gOut, dim3(128), 0, stream>>>(ao, wo, y);
}